// ConCare_14405320311493
// MI455X (gfx1250) — compile-verified
//
#include <hip/hip_runtime.h>
#include <math.h>

typedef float v2f __attribute__((ext_vector_type(2)));
typedef float v8f __attribute__((ext_vector_type(8)));
typedef unsigned int v4u __attribute__((ext_vector_type(4)));
typedef int v4i __attribute__((ext_vector_type(4)));
typedef int v8i __attribute__((ext_vector_type(8)));

#define Bb 64
#define Nf 76
#define Tt 128
#define Dd 128
#define AHh 8
#define NHEADS 4
#define DKh 32
#define DFFe 512
#define FAHe 128
#define MN (Bb*Nf)     // 4864 rows, 304 tiles of 16
#define RS 130         // LDS row stride for the H tile (TDM pad: 128 + 2)

// ---------------------------------------------------------------------------
// Stage 1: MissingAwareTemporalAttention, one block per (b,n).
// The 128x128 fp32 H tile (64KB) is DMA'd into LDS once by the Tensor Data
// Mover (pad = 2 DWORDs per 128-DWORD row -> stride 130, bank-friendly).
// dot[t] = H[t,:]·u with u = Wx @ (H[T-1,:] @ Wt); softmax over T; emb = a·H.
// H is read from HBM exactly once.
// ---------------------------------------------------------------------------
__global__ __launch_bounds__(128) void stage1_kernel(
    const float* __restrict__ H, const float* __restrict__ mask,
    const float* __restrict__ Wt, const float* __restrict__ Wx,
    const float* __restrict__ rate, const float* __restrict__ obs_bias,
    const float* __restrict__ miss_bias, float* __restrict__ emb)
{
  extern __shared__ float hs[];          // [Tt][RS] = 128*130 floats (66,560 B)
  __shared__ float q8[AHh];
  __shared__ float u[Dd];
  __shared__ float red[128];
  __shared__ float avec[Tt];

  const int bn  = blockIdx.x;
  const int tid = threadIdx.x;
  const float* Hp = H + (size_t)bn * (Tt*Dd);

  if (tid < 32) {   // wave 0 issues the TDM descriptor (uniform operands)
    const unsigned long long gaddr = (unsigned long long)(const void*)Hp;
    const unsigned ldsoff = (unsigned)(unsigned long long)(void*)hs;
    v4u g0;
    g0[0] = 1u;                                        // count=1, user mode
    g0[1] = ldsoff;                                    // lds_addr
    g0[2] = (unsigned)(gaddr & 0xFFFFFFFFull);         // global_addr[31:0]
    g0[3] = (unsigned)((gaddr >> 32) & 0x1FFFFFFull)   // global_addr[56:32]
            | 0x80000000u;                             // type=2 ("image")
    v8i g1;
    g1[0] = (int)0x03920000u;  // data_size=4B | pad_en | interval=128DW | amount=2DW
    g1[1] = (int)(128u << 16); // tensor_dim0[15:0]=128 (bits 79:64 -> next word = 0)
    g1[2] = (int)(128u << 16); // tensor_dim1[15:0]=128
    g1[3] = (int)(128u << 16); // tile_dim0=128
    g1[4] = 128;               // tile_dim1=128, tile_dim2=0
    g1[5] = 128;               // tensor_dim0_stride=128
    g1[6] = 0;                 // tensor_dim1_stride (unused, 2D tile)
    g1[7] = 0;
    const v4i gz4 = {0, 0, 0, 0};                      // groups 2/3: <=2D tensor
    const v8i gz8 = {0, 0, 0, 0, 0, 0, 0, 0};
    __builtin_amdgcn_tensor_load_to_lds(g0, g1, gz4, gz4, gz8, 0);
    __builtin_amdgcn_s_wait_tensorcnt(0);
  }
  __syncthreads();   // publish the tile to all 4 waves

  // q[a] = H[T-1,:] @ Wt   (8 values)
  if (tid < AHh) {
    float s = 0.f;
    const float* hl = &hs[(Tt-1)*RS];
    for (int d = 0; d < Dd; ++d) s += hl[d] * Wt[d*AHh + tid];
    q8[tid] = s;
  }
  __syncthreads();
  {  // u[d] = Wx[d,:] @ q
    float s = 0.f;
    for (int a = 0; a < AHh; ++a) s += Wx[tid*AHh + a] * q8[a];
    u[tid] = s;
  }
  __syncthreads();

  // dot[t]: thread t owns row t in LDS (stride 130 -> conflict-free per wave)
  float dot = 0.f;
  {
    const float* rowp = &hs[tid*RS];
    for (int d = 0; d < Dd; ++d) dot += rowp[d] * u[d];
  }

  const float sig   = 1.f/(1.f + expf(-dot));
  const float rt    = 1.f/(1.f + expf(-rate[0]));
  const float btime = (float)(Tt - tid);                // arange(T-1..0)+1
  const float denom = rt * (logf(2.72f + (1.0f - sig)) * btime);
  float e = sig / denom;
  e = e > 0.f ? e : 0.f;
  e += (mask[(size_t)bn*Tt + tid] > 0.5f) ? obs_bias[0] : miss_bias[0];

  red[tid] = e; __syncthreads();
  for (int s = 64; s > 0; s >>= 1) { if (tid < s) red[tid] = fmaxf(red[tid], red[tid+s]); __syncthreads(); }
  const float mx = red[0]; __syncthreads();
  const float ex = expf(e - mx);
  red[tid] = ex; __syncthreads();
  for (int s = 64; s > 0; s >>= 1) { if (tid < s) red[tid] += red[tid+s]; __syncthreads(); }
  avec[tid] = ex / red[0];
  __syncthreads();

  // emb[d] = sum_t a[t]*H[t,d]  (column read, conflict-free)
  float acc = 0.f;
  for (int t = 0; t < Tt; ++t) acc += avec[t] * hs[t*RS + tid];
  emb[(size_t)bn*Dd + tid] = acc;
}

// ---------------------------------------------------------------------------
// LayerNorm (torch-style: unbiased std, eps added to std). One block per row.
// ---------------------------------------------------------------------------
__global__ __launch_bounds__(128) void ln_kernel(const float* __restrict__ X,
    const float* __restrict__ g, const float* __restrict__ b,
    float* __restrict__ Y)
{
  __shared__ float red[128];
  const int row = blockIdx.x, d = threadIdx.x;
  const float x = X[(size_t)row*Dd + d];
  red[d] = x; __syncthreads();
  for (int s = 64; s > 0; s >>= 1) { if (d < s) red[d] += red[d+s]; __syncthreads(); }
  const float mean = red[0] * (1.0f/Dd); __syncthreads();
  const float c = x - mean;
  red[d] = c*c; __syncthreads();
  for (int s = 64; s > 0; s >>= 1) { if (d < s) red[d] += red[d+s]; __syncthreads(); }
  const float stdv = sqrtf(red[0] / (float)(Dd - 1));
  Y[(size_t)row*Dd + d] = g[d] * c / (stdv + 1e-7f) + b[d];
}

// ---------------------------------------------------------------------------
// fp32 GEMM via V_WMMA_F32_16X16X4_F32. C = [relu](A@W + bias) [+ R].
// A 16xK strip is staged once per block into LDS (stride K+2 -> conflict-free
// 16-lane reads); the k-loop is pure pointer-bump: 2 ds_load + 2 global_load
// + 1 wmma per step. Wave w owns tile (blockIdx.x, 4*blockIdx.y+w).
// ---------------------------------------------------------------------------
__global__ __launch_bounds__(128) void gemm_wmma_kernel(
    const float* __restrict__ A, int lda,
    const float* __restrict__ W, int ldw,
    const float* __restrict__ bias,
    const float* __restrict__ R,           // residual (ld = ldc) or nullptr
    float* __restrict__ C, int ldc,
    int K, int applyRelu)
{
  __shared__ float As[16*(DFFe+2)];        // max K = 512 -> 32,896 B
  const int tid  = threadIdx.x;
  const int lane = tid & 31;
  const int wv   = tid >> 5;
  const int m0   = blockIdx.x * 16;
  const int n0   = (blockIdx.y * 4 + wv) * 16;
  const int l15  = lane & 15;
  const int half = lane >> 4;
  const int rs   = K + 2;

  // stage A[16][K]: coalesced float4 global loads, scalar LDS stores (pad)
  for (int i = tid*4; i < 16*K; i += 128*4) {
    const int r = i / K, ccol = i % K;     // K % 4 == 0 -> no row crossing
    const float4 v4 = *(const float4*)(A + (size_t)(m0 + r)*lda + ccol);
    float* dst = &As[r*rs + ccol];
    dst[0]=v4.x; dst[1]=v4.y; dst[2]=v4.z; dst[3]=v4.w;
  }
  __syncthreads();

  const float* ap = &As[l15*rs + 2*half];
  const float* wp = W + (size_t)(2*half)*ldw + n0 + l15;
  const size_t wstep = (size_t)4*ldw;
  v8f acc = {0.f,0.f,0.f,0.f,0.f,0.f,0.f,0.f};
  for (int k = 0; k < K; k += 4) {
    v2f a;  a.x  = ap[0];  a.y  = ap[1];
    v2f bm; bm.x = wp[0];  bm.y = wp[ldw];
    acc = __builtin_amdgcn_wmma_f32_16x16x4_f32(false, a, false, bm,
                                                (short)0, acc, false, false);
    ap += 4;
    wp += wstep;
  }
  const float bval = bias ? bias[n0 + l15] : 0.f;
  #pragma unroll
  for (int r = 0; r < 8; ++r) {
    const int row = m0 + r + 8*half;
    const int col = n0 + l15;
    float v = acc[r] + bval;
    if (applyRelu) v = v > 0.f ? v : 0.f;
    if (R) v += R[(size_t)row*ldc + col];
    C[(size_t)row*ldc + col] = v;
  }
}

// ---------------------------------------------------------------------------
// MHA core: one block per (b, head). N=76, DK=32; scores kept in LDS.
// ---------------------------------------------------------------------------
__global__ __launch_bounds__(128) void attn_kernel(
    const float* __restrict__ Q, const float* __restrict__ Km,
    const float* __restrict__ V, float* __restrict__ ctx)
{
  __shared__ float ks[Nf][DKh+1];
  __shared__ float vs[Nf][DKh+1];
  __shared__ float sc[Nf][Nf+1];
  const int bh = blockIdx.x;
  const int b = bh >> 2, h = bh & 3;
  const int tid = threadIdx.x;
  for (int i = tid; i < Nf*DKh; i += 128) {
    const int n = i >> 5, c = i & 31;
    const size_t base = ((size_t)(b*Nf + n))*Dd + h*DKh + c;
    ks[n][c] = Km[base];
    vs[n][c] = V[base];
  }
  __syncthreads();
  if (tid < Nf) {
    const float scale = 0.17677669529663688f;  // 1/sqrt(32)
    float qr[DKh];
    const size_t qbase = ((size_t)(b*Nf + tid))*Dd + h*DKh;
    for (int c = 0; c < DKh; ++c) qr[c] = Q[qbase + c];
    float mx = -3.4e38f;
    for (int j = 0; j < Nf; ++j) {
      float s = 0.f;
      for (int c = 0; c < DKh; ++c) s += qr[c]*ks[j][c];
      s *= scale;
      sc[tid][j] = s;
      mx = fmaxf(mx, s);
    }
    float ssum = 0.f;
    for (int j = 0; j < Nf; ++j) { float p = expf(sc[tid][j] - mx); sc[tid][j] = p; ssum += p; }
    const float inv = 1.f/ssum;
    float outv[DKh];
    for (int c = 0; c < DKh; ++c) outv[c] = 0.f;
    for (int j = 0; j < Nf; ++j) {
      const float p = sc[tid][j]*inv;
      for (int c = 0; c < DKh; ++c) outv[c] += p*vs[j][c];
    }
    float* op = ctx + ((size_t)(b*Nf + tid))*Dd + h*DKh;
    for (int c = 0; c < DKh; ++c) op[c] = outv[c];
  }
}

__global__ void add_kernel(const float* __restrict__ a, const float* __restrict__ b,
                           float* __restrict__ c, int n)
{
  const int i = blockIdx.x*blockDim.x + threadIdx.x;
  if (i < n) c[i] = a[i] + b[i];
}

// ---------------------------------------------------------------------------
// DeCov: per feature n, cov = centered @ centered^T / (B-1) over batch.
// 0.5 * sum_{d!=e} cov[d,e]^2 accumulated via WMMA tiles (8x8 tiles, K=64).
// ---------------------------------------------------------------------------
__global__ __launch_bounds__(256) void decov_kernel(const float* __restrict__ mha,
                                                    float* __restrict__ out_decov)
{
  __shared__ float c[Dd][Bb+1];     // [128][65] -> conflict-free rows
  __shared__ float red[256];
  const int n = blockIdx.x;
  const int tid = threadIdx.x;
  for (int i = tid; i < Bb*Dd; i += 256) {
    const int b = i >> 7, d = i & 127;
    c[d][b] = mha[((size_t)(b*Nf + n))*Dd + d];
  }
  __syncthreads();
  if (tid < Dd) {
    float s = 0.f;
    for (int b = 0; b < Bb; ++b) s += c[tid][b];
    s *= (1.0f/Bb);
    for (int b = 0; b < Bb; ++b) c[tid][b] -= s;
  }
  __syncthreads();

  const int wave = tid >> 5, lane = tid & 31;
  const int l15 = lane & 15, half = lane >> 4;
  float local = 0.f;
  for (int tileIdx = wave; tileIdx < 64; tileIdx += 8) {
    const int ti = tileIdx >> 3, tj = tileIdx & 7;
    v8f acc = {0.f,0.f,0.f,0.f,0.f,0.f,0.f,0.f};
    for (int k = 0; k < Bb; k += 4) {
      const int ka = k + 2*half;
      v2f a;  a.x  = c[ti*16 + l15][ka+0]; a.y  = c[ti*16 + l15][ka+1];
      v2f bm; bm.x = c[tj*16 + l15][ka+0]; bm.y = c[tj*16 + l15][ka+1];
      acc = __builtin_amdgcn_wmma_f32_16x16x4_f32(false, a, false, bm,
                                                  (short)0, acc, false, false);
    }
    #pragma unroll
    for (int r = 0; r < 8; ++r) {
      const int drow = ti*16 + r + 8*half;
      const int dcol = tj*16 + l15;
      const float val = acc[r] * (1.0f/(Bb-1));
      if (drow != dcol) local += 0.5f*val*val;
    }
  }
  red[tid] = local; __syncthreads();
  for (int s = 128; s > 0; s >>= 1) { if (tid < s) red[tid] += red[tid+s]; __syncthreads(); }
  if (tid == 0) atomicAdd(out_decov, red[0]);
}

// ---------------------------------------------------------------------------
// FinalAttentionQKV pooling: one block per batch element.
// ---------------------------------------------------------------------------
__global__ __launch_bounds__(128) void finalpool_kernel(
    const float* __restrict__ fk, const float* __restrict__ fv,
    const float* __restrict__ fq, const float* __restrict__ fmask,
    const float* __restrict__ mscale, float* __restrict__ out)
{
  __shared__ float fqs[FAHe];
  __shared__ float red[128];
  __shared__ float alpha[Nf];
  const int b = blockIdx.x, tid = threadIdx.x;
  fqs[tid] = fq[(size_t)b*FAHe + tid];
  __syncthreads();
  float e = -3.4e38f;
  if (tid < Nf) {
    const float* kr = fk + ((size_t)(b*Nf + tid))*FAHe;
    float s = 0.f;
    for (int a = 0; a < FAHe; ++a) s += kr[a]*fqs[a];
    e = s + mscale[0]*fmask[(size_t)b*Nf + tid];
  }
  red[tid] = e; __syncthreads();
  for (int s = 64; s > 0; s >>= 1) { if (tid < s) red[tid] = fmaxf(red[tid], red[tid+s]); __syncthreads(); }
  const float mx = red[0]; __syncthreads();
  const float ex = (tid < Nf) ? expf(e - mx) : 0.f;
  red[tid] = ex; __syncthreads();
  for (int s = 64; s > 0; s >>= 1) { if (tid < s) red[tid] += red[tid+s]; __syncthreads(); }
  const float inv = 1.f/red[0];
  if (tid < Nf) alpha[tid] = ex*inv;
  __syncthreads();
  float acc = 0.f;
  for (int nn = 0; nn < Nf; ++nn) acc += alpha[nn]*fv[((size_t)(b*Nf + nn))*FAHe + tid];
  out[(size_t)b*FAHe + tid] = acc;
}

__global__ void zero_kernel(float* p) { *p = 0.f; }

// ---------------------------------------------------------------------------
extern "C" void kernel_launch(void* const* d_in, const int* in_sizes, int n_in,
                              void* d_out, int out_size, void* d_ws, size_t ws_size,
                              hipStream_t stream)
{
  (void)in_sizes; (void)n_in; (void)out_size; (void)ws_size;
  const float* H     = (const float*)d_in[0];
  const float* mask  = (const float*)d_in[1];
  const float* fmask = (const float*)d_in[2];
  const float* Wt    = (const float*)d_in[3];
  const float* Wx    = (const float*)d_in[4];
  const float* rate  = (const float*)d_in[5];
  const float* obsb  = (const float*)d_in[6];
  const float* missb = (const float*)d_in[7];
  const float* Wq = (const float*)d_in[8];  const float* bq = (const float*)d_in[9];
  const float* Wk = (const float*)d_in[10]; const float* bk = (const float*)d_in[11];
  const float* Wv = (const float*)d_in[12]; const float* bv = (const float*)d_in[13];
  const float* Wo = (const float*)d_in[14]; const float* bo = (const float*)d_in[15];
  const float* ln1a = (const float*)d_in[16]; const float* ln1b = (const float*)d_in[17];
  const float* ln2a = (const float*)d_in[18]; const float* ln2b = (const float*)d_in[19];
  const float* w1 = (const float*)d_in[20]; const float* b1 = (const float*)d_in[21];
  const float* w2 = (const float*)d_in[22]; const float* b2 = (const float*)d_in[23];
  const float* fWq = (const float*)d_in[24]; const float* fbq = (const float*)d_in[25];
  const float* fWk = (const float*)d_in[26]; const float* fbk = (const float*)d_in[27];
  const float* fWv = (const float*)d_in[28]; const float* fbv = (const float*)d_in[29];
  const float* mscale = (const float*)d_in[30];

  float* ws = (float*)d_ws;
  const size_t SZ = (size_t)MN * Dd;           // 622592 floats
  float* emb  = ws;                            // [MN,128]
  float* x0   = emb + SZ;                      // also reused as x1
  float* big  = x0 + SZ;                       // 4*SZ region
  float* qb   = big;
  float* kb   = big + SZ;
  float* vb   = big + 2*SZ;
  float* ctx  = big + 3*SZ;
  float* hbuf = big;                           // FFN hidden [MN,512] == 4*SZ
  float* mha  = big + 4*SZ;                    // also reused as x2
  float* x    = mha + SZ;
  float* fqb  = x + SZ;                        // [64,128]
  float* fkb  = big;                           // reuse after hbuf is dead
  float* fvb  = big + SZ;

  float* out = (float*)d_out;                  // [0..8191]=pooled, [8192]=decov

  // opt-in to >64KB dynamic LDS for stage1 (host-side, graph-capture safe)
  const int s1_lds = Tt * RS * (int)sizeof(float);   // 66,560 B
  (void)hipFuncSetAttribute((const void*)stage1_kernel,
                            hipFuncAttributeMaxDynamicSharedMemorySize, s1_lds);

  zero_kernel<<<1, 1, 0, stream>>>(out + Bb*FAHe);

  stage1_kernel<<<MN, 128, s1_lds, stream>>>(H, mask, Wt, Wx, rate, obsb, missb, emb);

  // Sublayer 1: pre-norm + MHA + residual
  ln_kernel<<<MN, 128, 0, stream>>>(emb, ln1a, ln1b, x0);
  dim3 g1(MN/16, Dd/64);
  gemm_wmma_kernel<<<g1, 128, 0, stream>>>(x0, Dd, Wq, Dd, bq, nullptr, qb, Dd, Dd, 0);
  gemm_wmma_kernel<<<g1, 128, 0, stream>>>(x0, Dd, Wk, Dd, bk, nullptr, kb, Dd, Dd, 0);
  gemm_wmma_kernel<<<g1, 128, 0, stream>>>(x0, Dd, Wv, Dd, bv, nullptr, vb, Dd, Dd, 0);
  attn_kernel<<<Bb*NHEADS, 128, 0, stream>>>(qb, kb, vb, ctx);
  gemm_wmma_kernel<<<g1, 128, 0, stream>>>(ctx, Dd, Wo, Dd, bo, nullptr, mha, Dd, Dd, 0);
  add_kernel<<<(MN*Dd)/256, 256, 0, stream>>>(emb, mha, x, MN*Dd);
  decov_kernel<<<Nf, 256, 0, stream>>>(mha, out + Bb*FAHe);

  // Sublayer 2: pre-norm + FFN + residual (x2 written into mha's slot)
  ln_kernel<<<MN, 128, 0, stream>>>(x, ln2a, ln2b, x0);
  dim3 g2(MN/16, DFFe/64);
  gemm_wmma_kernel<<<g2, 128, 0, stream>>>(x0, Dd, w1, DFFe, b1, nullptr, hbuf, DFFe, Dd, 1);
  gemm_wmma_kernel<<<g1, 128, 0, stream>>>(hbuf, DFFe, w2, Dd, b2, x, mha, Dd, DFFe, 0);
  float* x2 = mha;

  // FinalAttentionQKV ('mul')
  dim3 g3(Bb/16, FAHe/64);
  gemm_wmma_kernel<<<g3, 128, 0, stream>>>(x2 + (size_t)(Nf-1)*Dd, Nf*Dd,
                                           fWq, FAHe, fbq, nullptr, fqb, FAHe, Dd, 0);
  gemm_wmma_kernel<<<g1, 128, 0, stream>>>(x2, Dd, fWk, FAHe, fbk, nullptr, fkb, FAHe, Dd, 0);
  gemm_wmma_kernel<<<g1, 128, 0, stream>>>(x2, Dd, fWv, FAHe, fbv, nullptr, fvb, FAHe, Dd, 0);
  finalpool_kernel<<<Bb, 128, 0, stream>>>(fkb, fvb, fqb, fmask, mscale, out);
}